// Encoder_515396075689
// MI455X (gfx1250) — compile-verified
//
#include <hip/hip_runtime.h>
#include <math.h>

// Problem constants (from reference)
#define SEQ 633
#define IN  28
#define H1  14
#define H2  7
#define G1N  (4 * H1)     // 56 gate pre-activations, layer 1
#define G2N  (4 * H2)     // 28 gate pre-activations, layer 2
#define MPAD 640          // SEQ padded to 40 tiles of 16 (g1 row padding)
#define WDOT (H1 + H2)    // 21: unified per-lane dot length

typedef float v2f __attribute__((ext_vector_type(2)));
typedef float v8f __attribute__((ext_vector_type(8)));

static __device__ __forceinline__ int imin(int a, int b) { return a < b ? a : b; }

// ---------------------------------------------------------------------------
// Kernel 1: parallel input projection for layer 1 using f32 WMMA.
//   G1[t][n] = sum_k x[t][k] * w_ih1[n][k] + (b_ih1[n] + b_hh1[n])
// M = 633 (padded to 640 = 40 tiles), N = 56 (padded to 64 = 4 tiles),
// K = 28 (7 chunks of 4). One wave per 16x16 tile, 7 chained
// v_wmma_f32_16x16x4_f32, bias folded into the C accumulator.
// Address clamping (not predication) keeps EXEC all-ones for WMMA; g1 is
// row-padded to 640 so stores need only a single per-lane column guard.
// ---------------------------------------------------------------------------
__global__ __launch_bounds__(32) void lstm1_input_gemm(
    const float* __restrict__ x,      // [SEQ][IN]
    const float* __restrict__ w_ih1,  // [G1N][IN]
    const float* __restrict__ b_ih1,  // [G1N]
    const float* __restrict__ b_hh1,  // [G1N]
    float* __restrict__ g1)           // [MPAD][G1N]
{
    const int lane  = threadIdx.x;    // 0..31
    const int mlane = lane & 15;      // M (for A) / N (for B,C,D) within tile
    const int hi    = lane >> 4;      // lane half
    const int row0  = blockIdx.x * 16;
    const int col0  = blockIdx.y * 16;

    const int am = imin(row0 + mlane, SEQ - 1);   // clamped A row (pad rows duplicate)
    const int n  = col0 + mlane;                  // this lane's B/C/D column
    const int nc = imin(n, G1N - 1);              // clamped weight/bias row

    const float bias = b_ih1[nc] + b_hh1[nc];
    v8f c;
#pragma unroll
    for (int i = 0; i < 8; ++i) c[i] = bias;

#pragma unroll
    for (int kb = 0; kb < IN; kb += 4) {
        // A 16x4 tile: lane l, vgpr v -> A[M=l%16][K = kb + v + 2*(l>>4)]
        const int ak = kb + 2 * hi;
        v2f a;
        a.x = x[am * IN + ak + 0];
        a.y = x[am * IN + ak + 1];
        // B 4x16 tile, B[k][n] = w_ih1[n][k]: lane l, vgpr v -> K = kb+v+2*(l>>4), N = l%16
        v2f b;
        b.x = w_ih1[nc * IN + ak + 0];
        b.y = w_ih1[nc * IN + ak + 1];
        c = __builtin_amdgcn_wmma_f32_16x16x4_f32(false, a, false, b, (short)0, c,
                                                  false, false);
    }

    // D 16x16: vgpr v, lane l -> M = v + 8*(l>>4), N = l%16. Rows unguarded (padded).
    if (n < G1N) {
#pragma unroll
        for (int v = 0; v < 8; ++v)
            g1[(row0 + v + 8 * hi) * G1N + n] = c[v];
    }
}

// ---------------------------------------------------------------------------
// Kernel 2: sequential recurrence, one wave32 workgroup, software-pipelined.
// Layer-2 trails layer-1 by one timestep so both run concurrently in
// DIFFERENT LANES of one branchless instruction stream:
//   - lanes 0..13 : layer-1 unit `lane`  (weights [W_hh1 row | 0x7])
//   - lanes 16..22: layer-2 unit lane-16 (weights [W_ih2 row | W_hh2 row])
// Every lane: 4 FMA-dots of length 21 vs shared v = [h1 | h2] (LDS
// broadcast), per-lane weights in registers, cell state in a register.
// g1 rows are register-double-buffered one step ahead + prefetched two ahead.
// ---------------------------------------------------------------------------
__device__ __forceinline__ float sigm(float v) {
    return 1.0f / (1.0f + __expf(-v));
}
__device__ __forceinline__ float tanh_fast(float v) {
    return 1.0f - 2.0f / (__expf(2.0f * v) + 1.0f);
}

__global__ __launch_bounds__(32) void lstm_recurrent(
    const float* __restrict__ g1,     // [MPAD][G1N] precomputed x@Wih1^T + b1
    const float* __restrict__ w_hh1,  // [G1N][H1]
    const float* __restrict__ w_ih2,  // [G2N][H1]
    const float* __restrict__ w_hh2,  // [G2N][H2]
    const float* __restrict__ b_ih2,  // [G2N]
    const float* __restrict__ b_hh2,  // [G2N]
    float* __restrict__ out)          // [SEQ][H2]
{
    __shared__ __align__(16) float v[32];  // [0..13]=h1, [14..20]=h2, rest dummy

    const int  lane = threadIdx.x;
    const bool isL1 = lane < H1;
    const bool isL2 = (lane >= 16) && (lane < 16 + H2);
    const int  j2   = lane - 16;

    // unique LDS slot per lane (active lanes map to state slots, rest dummy)
    int widx;
    if      (lane < H1)      widx = lane;      // h1 slots 0..13
    else if (lane < 16)      widx = lane + 7;  // 14,15 -> dummy 21,22
    else if (lane < 16 + H2) widx = lane - 2;  // 16..22 -> h2 slots 14..20
    else                     widx = lane;      // 23..31 dummy

    // per-lane weight rows in registers; zero for dummy lanes / layer-1 tail
    float wreg[4][WDOT];
#pragma unroll
    for (int g = 0; g < 4; ++g)
#pragma unroll
        for (int k = 0; k < WDOT; ++k) wreg[g][k] = 0.0f;

    if (isL1) {
#pragma unroll
        for (int g = 0; g < 4; ++g)
#pragma unroll
            for (int k = 0; k < H1; ++k)
                wreg[g][k] = w_hh1[(g * H1 + lane) * H1 + k];
    }
    if (isL2) {
#pragma unroll
        for (int g = 0; g < 4; ++g) {
#pragma unroll
            for (int k = 0; k < H1; ++k)
                wreg[g][k] = w_ih2[(g * H2 + j2) * H1 + k];
#pragma unroll
            for (int k = 0; k < H2; ++k)
                wreg[g][H1 + k] = w_hh2[(g * H2 + j2) * H2 + k];
        }
    }

    // layer-2 fused bias in registers (layer-1 lanes take g1 instead)
    float bias[4] = {0.0f, 0.0f, 0.0f, 0.0f};
    if (isL2) {
#pragma unroll
        for (int g = 0; g < 4; ++g)
            bias[g] = b_ih2[g * H2 + j2] + b_hh2[g * H2 + j2];
    }

    // g1 gather columns for layer-1 lanes (others clamp to 0: harmless load)
    int gidx[4];
#pragma unroll
    for (int g = 0; g < 4; ++g) gidx[g] = isL1 ? (g * H1 + lane) : 0;

    v[lane] = 0.0f;
    __syncthreads();

    // preload g1 row 0
    float gv[4];
#pragma unroll
    for (int g = 0; g < 4; ++g) gv[g] = g1[gidx[g]];

    float cst = 0.0f;  // this lane's cell state

    for (int t = 0; t <= SEQ; ++t) {   // iteration t: layer1 step t, layer2 step t-1
        // double-buffer next g1 row; prefetch one further ahead
        const int tn = imin(t + 1, SEQ - 1);
        float gnext[4];
#pragma unroll
        for (int g = 0; g < 4; ++g) gnext[g] = g1[tn * G1N + gidx[g]];
        __builtin_prefetch(&g1[imin(t + 2, SEQ - 1) * G1N], 0, 0);

        // broadcast shared state into registers (contiguous -> b128 LDS loads)
        float vv[WDOT];
#pragma unroll
        for (int k = 0; k < WDOT; ++k) vv[k] = v[k];

        float a0 = isL1 ? gv[0] : bias[0];
        float a1 = isL1 ? gv[1] : bias[1];
        float a2 = isL1 ? gv[2] : bias[2];
        float a3 = isL1 ? gv[3] : bias[3];
#pragma unroll
        for (int k = 0; k < WDOT; ++k) {
            a0 = fmaf(wreg[0][k], vv[k], a0);
            a1 = fmaf(wreg[1][k], vv[k], a1);
            a2 = fmaf(wreg[2][k], vv[k], a2);
            a3 = fmaf(wreg[3][k], vv[k], a3);
        }
        float cc = sigm(a1) * cst + sigm(a0) * tanh_fast(a2);
        float hh = sigm(a3) * tanh_fast(cc);
        if (t == 0 && !isL1) { cc = 0.0f; hh = 0.0f; }  // layer-2 pipeline warm-up
        cst = cc;

        __syncthreads();           // all v reads complete (1-wave WG: near-free)
        v[widx] = hh;
        __syncthreads();           // new state visible

        if (isL2 && t >= 1) out[(t - 1) * H2 + j2] = hh;

#pragma unroll
        for (int g = 0; g < 4; ++g) gv[g] = gnext[g];
    }
}

// ---------------------------------------------------------------------------
extern "C" void kernel_launch(void* const* d_in, const int* in_sizes, int n_in,
                              void* d_out, int out_size, void* d_ws, size_t ws_size,
                              hipStream_t stream) {
    (void)in_sizes; (void)n_in; (void)out_size; (void)ws_size;
    const float* x     = (const float*)d_in[0];
    const float* w_ih1 = (const float*)d_in[1];
    const float* w_hh1 = (const float*)d_in[2];
    const float* b_ih1 = (const float*)d_in[3];
    const float* b_hh1 = (const float*)d_in[4];
    const float* w_ih2 = (const float*)d_in[5];
    const float* w_hh2 = (const float*)d_in[6];
    const float* b_ih2 = (const float*)d_in[7];
    const float* b_hh2 = (const float*)d_in[8];
    float* out = (float*)d_out;
    float* g1  = (float*)d_ws;            // [MPAD][G1N] = 640*56 floats (~140 KB)

    dim3 grid1(MPAD / 16, (G1N + 15) / 16);   // 40 x 4 tiles
    lstm1_input_gemm<<<grid1, 32, 0, stream>>>(x, w_ih1, b_ih1, b_hh1, g1);
    lstm_recurrent<<<1, 32, 0, stream>>>(g1, w_hh1, w_ih2, w_hh2, b_ih2, b_hh2, out);
}